// MultiBoxesLoss_57904749084971
// MI455X (gfx1250) — compile-verified
//
#include <hip/hip_runtime.h>
#include <math.h>

// SSD MultiBox loss for MI455X (gfx1250, wave32).
// B=64, P=8732, C=81, G=32 (derived at launch from in_sizes).
// Memory-bound: ~195MB total traffic -> ~8.4us floor @ 23.3 TB/s.

#define THRESH 0.5f
#define NEGPOS 3
#define MAXP   8736   // LDS sizing for k_match (P=8732)

typedef float v2f __attribute__((ext_vector_type(2)));
typedef float v8f __attribute__((ext_vector_type(8)));

#if __has_builtin(__builtin_amdgcn_global_load_async_to_lds_b128) && \
    __has_builtin(__builtin_amdgcn_s_wait_asynccnt)
#define HAVE_ASYNC_LDS 1
// Builtin wants pointers to 16B int vectors (per hipcc diagnostic), in the
// global (AS1) / LDS (AS3) address spaces.
typedef int vi4 __attribute__((vector_size(4 * sizeof(int))));
typedef __attribute__((address_space(1))) vi4  g_vi4;
typedef __attribute__((address_space(3))) vi4  l_vi4;
typedef __attribute__((address_space(1))) char g_char;
typedef __attribute__((address_space(3))) char l_char;
#else
#define HAVE_ASYNC_LDS 0
#endif

__device__ __forceinline__ float waveSum(float v) {
    #pragma unroll
    for (int off = 16; off; off >>= 1) v += __shfl_xor(v, off, 32);
    return v;
}
__device__ __forceinline__ float waveMax(float v) {
    #pragma unroll
    for (int off = 16; off; off >>= 1) v = fmaxf(v, __shfl_xor(v, off, 32));
    return v;
}
__device__ __forceinline__ int waveSumI(int v) {
    #pragma unroll
    for (int off = 16; off; off >>= 1) v += __shfl_xor(v, off, 32);
    return v;
}

__device__ __forceinline__ float iou_xy(float ax0, float ay0, float ax1, float ay1,
                                        float bx0, float by0, float bx1, float by1) {
    float lx = fmaxf(ax0, bx0), ly = fmaxf(ay0, by0);
    float rx = fminf(ax1, bx1), ry = fminf(ay1, by1);
    float w = fmaxf(rx - lx, 0.0f), h = fmaxf(ry - ly, 0.0f);
    float inter = w * h;
    float aa = (ax1 - ax0) * (ay1 - ay0);
    float ab = (bx1 - bx0) * (by1 - by0);
    return inter / (aa + ab - inter);
}

// ---------------------------------------------------------------------------
// Kernel 1: per-image prior<->GT matching. One block (256 thr) per image.
// ---------------------------------------------------------------------------
__global__ __launch_bounds__(256) void k_match(
    const float* __restrict__ boxes,   // [B,G,4] xy
    const float* __restrict__ priors,  // [P,4]  cxcy
    int*  __restrict__ matchIdx,       // [B,P]  gt index or -1
    int*  __restrict__ nPos,           // [B]
    int B, int G, int P)
{
    __shared__ float         ovSh[MAXP];
    __shared__ unsigned char idxSh[MAXP];
    __shared__ float gt[32 * 4];
    __shared__ int   bestP[32];
    __shared__ int   waveCnt[8];

    const int b   = blockIdx.x;
    const int tid = threadIdx.x;
    const int wave = tid >> 5, lane = tid & 31;

    if (tid < G * 4) gt[tid] = boxes[(size_t)b * G * 4 + tid];
    __syncthreads();

    // Phase 1: per-prior best GT (first-max tie-break, ascending g).
    for (int p = tid; p < P; p += blockDim.x) {
        float pcx = priors[p * 4 + 0], pcy = priors[p * 4 + 1];
        float pw  = priors[p * 4 + 2], ph  = priors[p * 4 + 3];
        float px0 = pcx - pw * 0.5f, py0 = pcy - ph * 0.5f;
        float px1 = pcx + pw * 0.5f, py1 = pcy + ph * 0.5f;
        float best = -1.0f; int bi = 0;
        for (int g = 0; g < G; ++g) {
            float o = iou_xy(gt[g*4], gt[g*4+1], gt[g*4+2], gt[g*4+3], px0, py0, px1, py1);
            if (o > best) { best = o; bi = g; }
        }
        ovSh[p] = best;
        idxSh[p] = (unsigned char)bi;
    }
    __syncthreads();

    // Phase 2: per-GT best prior (argmax over P). Each wave owns G/8 GTs.
    int gPerWave = (G + 7) / 8;
    int gBeg = wave * gPerWave;
    int gEnd = gBeg + gPerWave; if (gEnd > G) gEnd = G;
    for (int gi = gBeg; gi < gEnd; ++gi) {
        float ax0 = gt[gi*4], ay0 = gt[gi*4+1], ax1 = gt[gi*4+2], ay1 = gt[gi*4+3];
        float best = -1.0f; int bi = 0;
        for (int p = lane; p < P; p += 32) {
            float pcx = priors[p*4+0], pcy = priors[p*4+1];
            float pw  = priors[p*4+2], ph  = priors[p*4+3];
            float o = iou_xy(ax0, ay0, ax1, ay1,
                             pcx - pw*0.5f, pcy - ph*0.5f, pcx + pw*0.5f, pcy + ph*0.5f);
            if (o > best) { best = o; bi = p; }
        }
        #pragma unroll
        for (int off = 16; off; off >>= 1) {
            float o2 = __shfl_xor(best, off, 32);
            int   i2 = __shfl_xor(bi,   off, 32);
            if (o2 > best || (o2 == best && i2 < bi)) { best = o2; bi = i2; }
        }
        if (lane == 0) bestP[gi] = bi;
    }
    __syncthreads();

    // Phase 3: force-match each GT's best prior (sequential, last-wins).
    if (tid == 0) {
        for (int g = 0; g < G; ++g) {
            int p = bestP[g];
            idxSh[p] = (unsigned char)g;
            ovSh[p]  = 1.0f;
        }
    }
    __syncthreads();

    // Phase 4: emit match index (fg labels are all >=1, so pos <=> ov>=0.5).
    int cnt = 0;
    for (int p = tid; p < P; p += blockDim.x) {
        bool pos = ovSh[p] >= THRESH;
        matchIdx[(size_t)b * P + p] = pos ? (int)idxSh[p] : -1;
        cnt += pos ? 1 : 0;
    }
    cnt = waveSumI(cnt);
    if (lane == 0) waveCnt[wave] = cnt;
    __syncthreads();
    if (tid == 0) {
        int t = 0;
        for (int w = 0; w < 8; ++w) t += waveCnt[w];
        nPos[b] = t;
    }
}

// ---------------------------------------------------------------------------
// Kernel 2: streaming CE + fused smooth-L1. One wave per (b,p) row of C=81.
// ---------------------------------------------------------------------------
__global__ __launch_bounds__(256) void k_ce(
    const float* __restrict__ predBoxes, // [B,P,4]
    const float* __restrict__ logits,    // [B,P,C]
    const float* __restrict__ boxes,     // [B,G,4]
    const int*   __restrict__ labels,    // [B,G]
    const float* __restrict__ priors,    // [P,4]
    const int*   __restrict__ matchIdx,  // [B,P]
    float* __restrict__ ceNeg,           // [B,P]
    float* __restrict__ posCeP,          // [gridDim.x]
    float* __restrict__ sl1P,            // [gridDim.x]
    int B, int G, int P, int C)
{
    __shared__ float wCe[8], wSl[8];
    const int wave = threadIdx.x >> 5, lane = threadIdx.x & 31;
    const int wid  = blockIdx.x * (blockDim.x >> 5) + wave;
    const int nW   = gridDim.x * (blockDim.x >> 5);
    const long total = (long)B * P;

    float accCe = 0.0f, accSl = 0.0f;

    for (long r = wid; r < total; r += nW) {
        const int b = (int)(r / P);
        const int p = (int)(r % P);
        const float* row = logits + (size_t)r * C;

        // CDNA5 prefetch of next row this wave will touch (global_prefetch_b8)
        if (r + nW < total) __builtin_prefetch(row + (size_t)nW * C + lane * 8, 0, 0);

        const int m   = matchIdx[r];
        const int tgt = (m < 0) ? 0 : labels[b * G + m];

        float x0 = (lane      < C) ? row[lane]      : -INFINITY;
        float x1 = (lane + 32 < C) ? row[lane + 32] : -INFINITY;
        float x2 = (lane + 64 < C) ? row[lane + 64] : -INFINITY;
        float mx = waveMax(fmaxf(x0, fmaxf(x1, x2)));
        float e  = ((lane      < C) ? expf(x0 - mx) : 0.0f)
                 + ((lane + 32 < C) ? expf(x1 - mx) : 0.0f)
                 + ((lane + 64 < C) ? expf(x2 - mx) : 0.0f);
        float se = waveSum(e);
        float tv = 0.0f;
        if (lane      == tgt) tv = x0;
        if (lane + 32 == tgt) tv = x1;
        if (lane + 64 == tgt) tv = x2;
        float xt = waveSum(tv);               // exactly one lane nonzero
        float ce = (mx + logf(se)) - xt;

        if (lane == 0) ceNeg[r] = (m >= 0) ? 0.0f : ce;

        if (m >= 0) {
            if (lane == 0) accCe += ce;
            if (lane < 4) {
                const float* gb = boxes + ((size_t)b * G + m) * 4;
                float b0 = gb[0], b1 = gb[1], b2 = gb[2], b3 = gb[3];
                float pc0 = priors[p*4+0], pc1 = priors[p*4+1];
                float pc2 = priors[p*4+2], pc3 = priors[p*4+3];
                float t;
                if      (lane == 0) t = (0.5f * (b0 + b2) - pc0) * 10.0f / pc2;
                else if (lane == 1) t = (0.5f * (b1 + b3) - pc1) * 10.0f / pc3;
                else if (lane == 2) t = logf((b2 - b0) / pc2) * 5.0f;
                else                t = logf((b3 - b1) / pc3) * 5.0f;
                float d  = predBoxes[(size_t)r * 4 + lane] - t;
                float ad = fabsf(d);
                accSl += (ad < 1.0f) ? 0.5f * d * d : ad - 0.5f;
            }
        }
    }

    accCe = waveSum(accCe);
    accSl = waveSum(accSl);
    if (lane == 0) { wCe[wave] = accCe; wSl[wave] = accSl; }
    __syncthreads();
    if (threadIdx.x == 0) {
        float c = 0.0f, s = 0.0f;
        for (int w = 0; w < (int)(blockDim.x >> 5); ++w) { c += wCe[w]; s += wSl[w]; }
        posCeP[blockIdx.x] = c;
        sl1P[blockIdx.x]   = s;
    }
}

// ---------------------------------------------------------------------------
// Kernel 3: per-image hard-negative mining. The per-image ce_neg slice
// (34,928B, 16B-aligned) is staged into LDS with GLOBAL_LOAD_ASYNC_TO_LDS_B128
// (CDNA5 async path, tracked on ASYNCcnt) when available, then a descending
// bitonic sort runs in LDS and ranks < min(3*n_pos, P) are summed.
// One 1024-thread block per image.
// ---------------------------------------------------------------------------
__global__ __launch_bounds__(1024) void k_topk(
    const float* __restrict__ ceNeg,  // [B,P]
    const int*   __restrict__ nPos,   // [B]
    float* __restrict__ hardNeg,      // [B]
    int P, int nsort)
{
    extern __shared__ float s[];
    const int b = blockIdx.x, tid = threadIdx.x, T = blockDim.x;
    const float* src = ceNeg + (size_t)b * P;

#if HAVE_ASYNC_LDS
    if ((P & 3) == 0) {
        // per-lane async copy: global cache -> LDS, no VGPR round trip
        g_char* g = (g_char*)(uintptr_t)src;
        l_char* l = (l_char*)(unsigned)(uintptr_t)(void*)s;
        const int nchunk = P >> 2;            // 16B chunks
        for (int i = tid; i < nchunk; i += T) {
            __builtin_amdgcn_global_load_async_to_lds_b128(
                (g_vi4*)(g + (size_t)i * 16), (l_vi4*)(l + i * 16),
                /*offset=*/0, /*cpol=*/0);
        }
        for (int i = P + tid; i < nsort; i += T) s[i] = -1.0f;  // pad sorts last
        __builtin_amdgcn_s_wait_asynccnt(0);
    } else
#endif
    {
        for (int i = tid; i < nsort; i += T)
            s[i] = (i < P) ? src[i] : -1.0f;  // ce>=0, pad sorts last
    }
    __syncthreads();

    for (int k = 2; k <= nsort; k <<= 1) {
        for (int j = k >> 1; j > 0; j >>= 1) {
            for (int i = tid; i < nsort; i += T) {
                int ixj = i ^ j;
                if (ixj > i) {
                    bool desc = ((i & k) == 0);
                    float a = s[i], c = s[ixj];
                    if (desc ? (a < c) : (a > c)) { s[i] = c; s[ixj] = a; }
                }
            }
            __syncthreads();
        }
    }

    int K  = NEGPOS * nPos[b];
    int Kc = (K < P) ? K : P;
    float part = 0.0f;
    for (int i = tid; i < Kc; i += T) part += s[i];
    __syncthreads();
    s[tid] = part;
    __syncthreads();
    for (int off = T >> 1; off; off >>= 1) {
        if (tid < off) s[tid] += s[tid + off];
        __syncthreads();
    }
    if (tid == 0) hardNeg[b] = s[0];
}

// ---------------------------------------------------------------------------
// Kernel 4: final fixed-order reduction. Single wave. The 64 per-image
// hard-neg sums are reduced with V_WMMA_F32_16X16X4_F32 (A=values, B=ones):
// D[m,n] = sum_k A[m,k], full f32 FMA precision.
// ---------------------------------------------------------------------------
__global__ __launch_bounds__(32) void k_final(
    const float* __restrict__ posCeP, const float* __restrict__ sl1P, int nb,
    const int*   __restrict__ nPos,   const float* __restrict__ hardNeg, int B,
    float* __restrict__ out)
{
    const int lane = threadIdx.x;

    float c = 0.0f, sl = 0.0f;
    for (int i = lane; i < nb; i += 32) { c += posCeP[i]; sl += sl1P[i]; }
    c  = waveSum(c);
    sl = waveSum(sl);

    int np = 0;
    for (int i = lane; i < B; i += 32) np += nPos[i];
    np = waveSumI(np);

    // A-matrix 16x4 f32 lane layout (ISA 7.12.2): lane<16 holds (M=lane,K=0/1),
    // lane>=16 holds (M=lane-16,K=2/3). Flat value j = M*4+K.
    int j0, j1;
    if (lane < 16) { j0 = 4 * lane;            j1 = 4 * lane + 1;            }
    else           { j0 = 4 * (lane - 16) + 2; j1 = 4 * (lane - 16) + 3;     }
    v2f a;
    a.x = (j0 < B) ? hardNeg[j0] : 0.0f;
    a.y = (j1 < B) ? hardNeg[j1] : 0.0f;
    v2f onesB; onesB.x = 1.0f; onesB.y = 1.0f;
    v8f acc = {0.f, 0.f, 0.f, 0.f, 0.f, 0.f, 0.f, 0.f};
    acc = __builtin_amdgcn_wmma_f32_16x16x4_f32(
        /*neg_a=*/false, a, /*neg_b=*/false, onesB,
        /*c_mod=*/(short)0, acc, /*reuse_a=*/false, /*reuse_b=*/false);
    // lanes 0-15 hold row-sums of rows 0-7 in acc[0..7]; lanes 16-31 rows 8-15.
    float half = acc[0] + acc[1] + acc[2] + acc[3] + acc[4] + acc[5] + acc[6] + acc[7];
    float hn = half + __shfl_xor(half, 16, 32);
    // tail if B > 64 (not hit for this problem, kept for generality)
    float ex = 0.0f;
    for (int i = 64 + lane; i < B; i += 32) ex += hardNeg[i];
    hn += waveSum(ex);

    if (lane == 0) {
        float npf = (float)np;
        float clsLoss = (hn + c) / npf;
        float boxLoss = sl / (npf * 4.0f);
        out[0] = clsLoss + boxLoss;
    }
}

// ---------------------------------------------------------------------------
extern "C" void kernel_launch(void* const* d_in, const int* in_sizes, int n_in,
                              void* d_out, int out_size, void* d_ws, size_t ws_size,
                              hipStream_t stream)
{
    (void)n_in; (void)out_size; (void)ws_size;
    const float* predBoxes = (const float*)d_in[0];
    const float* logits    = (const float*)d_in[1];
    const float* boxes     = (const float*)d_in[2];
    const int*   labels    = (const int*)d_in[3];
    const float* priors    = (const float*)d_in[4];
    float* out = (float*)d_out;

    const int P = in_sizes[4] / 4;
    const int B = in_sizes[0] / (P * 4);
    const int C = in_sizes[1] / (B * P);
    const int G = in_sizes[3] / B;
    const long BP = (long)B * P;

    // workspace layout (all fully written each call; no cross-call state)
    char* w = (char*)d_ws;
    float* ceNeg    = (float*)w; w += BP * sizeof(float);
    int*   matchIdx = (int*)w;   w += BP * sizeof(int);
    int*   nPos     = (int*)w;   w += ((B + 31) / 32 * 32) * sizeof(int);
    float* hardNeg  = (float*)w; w += ((B + 31) / 32 * 32) * sizeof(float);
    const int NB2 = 1024;
    float* posCeP   = (float*)w; w += NB2 * sizeof(float);
    float* sl1P     = (float*)w; w += NB2 * sizeof(float);

    k_match<<<B, 256, 0, stream>>>(boxes, priors, matchIdx, nPos, B, G, P);
    k_ce<<<NB2, 256, 0, stream>>>(predBoxes, logits, boxes, labels, priors,
                                  matchIdx, ceNeg, posCeP, sl1P, B, G, P, C);
    int nsort = 1; while (nsort < P) nsort <<= 1;
    k_topk<<<B, 1024, (size_t)nsort * sizeof(float), stream>>>(ceNeg, nPos, hardNeg, P, nsort);
    k_final<<<1, 32, 0, stream>>>(posCeP, sl1P, NB2, nPos, hardNeg, B, out);
}